// CARAFENaive_28063316312234
// MI455X (gfx1250) — compile-verified
//
#include <hip/hip_runtime.h>

typedef float v2f __attribute__((ext_vector_type(2)));
typedef float v8f __attribute__((ext_vector_type(8)));

#define H_   64
#define W_   64
#define OH   128
#define OW   128
#define CTOT 256
#define CG   64
#define K2   25
#define NG   4
#define FSTR 98   // padded channel stride for feature LDS [c][y*12+x] (96+2)

// Block: 256 threads = 8 waves. Block owns (n, g, 8x4 source super-tile).
// Wave owns a 2x2 source tile -> 16 output pixels (4x4). K = 6x6 = 36 union
// neighborhood -> 9 chained V_WMMA_F32_16X16X4_F32 per 16-channel tile.
__global__ __launch_bounds__(256) void carafe_wmma(
    const float* __restrict__ feat, const float* __restrict__ mask,
    float* __restrict__ out)
{
  __shared__ float sF[CG * FSTR];   // features [c][y][x], 25088 B
  __shared__ float sM[K2 * 128];    // masks [tap][ry*16+rx], 12800 B

  const int tid = threadIdx.x;
  const int n   = blockIdx.z;
  const int g   = blockIdx.y;
  const int stx = blockIdx.x & 7;   // W/8 super-tiles across
  const int sty = blockIdx.x >> 3;  // H/4 super-tiles down
  const int x0  = stx * 8;
  const int y0  = sty * 4;

  // ---- stage features[n, g*64+c, y0-2..y0+5, x0-2..x0+9] -> sF[c][y][x]
  const float* fbase = feat + ((size_t)n * CTOT + g * CG) * (H_ * W_);
  #pragma unroll 1
  for (int e = tid; e < CG * 96; e += 256) {
    int c  = e / 96;
    int r  = e - c * 96;
    int yy = r / 12;
    int xx = r - yy * 12;
    int gy = y0 - 2 + yy;
    int gx = x0 - 2 + xx;
    float v = 0.0f;
    if (gy >= 0 && gy < H_ && gx >= 0 && gx < W_)
      v = fbase[c * (H_ * W_) + gy * W_ + gx];
    sF[c * FSTR + r] = v;
  }

  // ---- stage masks[n, g*25+ch, 2*y0+ry, 2*x0+rx] -> sM[ch][ry*16+rx] (coalesced)
  const float* mb = mask + ((size_t)n * (K2 * NG) + g * K2) * (OH * OW)
                  + (2 * y0) * OW + 2 * x0;
  #pragma unroll 1
  for (int e = tid; e < K2 * 128; e += 256) {
    int ch = e >> 7;
    int r  = e & 127;
    int ry = r >> 4;
    int rx = r & 15;
    sM[e] = mb[(size_t)ch * (OH * OW) + ry * OW + rx];
  }
  __syncthreads();

  const int wave = tid >> 5;
  const int lane = tid & 31;
  const int tx = wave & 3, ty = wave >> 2;   // 4 x 2 tiles of 2x2 sources
  const int h0 = y0 + 2 * ty;
  const int w0 = x0 + 2 * tx;

  // A-operand lane roles per ISA 16x4 f32 layout
  const int M    = lane & 15;                // output pixel in 4x4 block
  const int half = lane >> 4;                // K-halves {0,1} vs {2,3}
  const int myo = M >> 2, mxo = M & 3;
  const int ti = myo >> 1, tj = mxo >> 1;    // owning source pixel in 2x2 tile
  const int lc = lane & 15;                  // B/D channel lane (N)
  const int mpos = (4 * ty + myo) * 16 + (4 * tx + mxo);  // block-local out pixel

  // Gather A (mask taps -> 36-slot union-K) from LDS; record B row offsets.
  v2f a[9];
  int fo[9];
  #pragma unroll
  for (int kk = 0; kk < 9; ++kk) {
    #pragma unroll
    for (int j = 0; j < 2; ++j) {
      const int Kl = 4 * kk + j;       // K for half==0 (compile-time)
      const int Kh = Kl + 2;           // K for half==1 (compile-time)
      int dy = half ? (Kh / 6) : (Kl / 6);   // cndmask of two constants
      int dx = half ? (Kh % 6) : (Kl % 6);
      int di = dy - ti, dj = dx - tj;
      float av = 0.0f;
      if (di >= 0 && di < 5 && dj >= 0 && dj < 5)
        av = sM[(di * 5 + dj) * 128 + mpos];
      a[kk][j] = av;
      if (j == 0)  // pair (K,K+1) is horizontally adjacent: one b64 row offset
        fo[kk] = (2 * ty + dy) * 12 + (2 * tx + dx);
    }
  }

  float* obase = out + ((size_t)n * CTOT + g * CG) * (OH * OW);

  #pragma unroll
  for (int ct = 0; ct < 4; ++ct) {           // 4 channel tiles of 16
    v8f acc = {};
    #pragma unroll
    for (int kk = 0; kk < 9; ++kk) {
      // aligned contiguous pair -> single ds_load_b64; ct offset is a DS imm
      v2f b = *(const v2f*)&sF[lc * FSTR + fo[kk] + ct * (16 * FSTR)];
      acc = __builtin_amdgcn_wmma_f32_16x16x4_f32(
          false, a[kk], false, b, (short)0, acc, false, false);
    }
    // D layout: acc[0..3] = row (2h0+2*half), cols 2w0..+3 ; acc[4..7] next row
    float* op = obase + (size_t)(ct * 16 + lc) * (OH * OW);
    const int ryo = 2 * h0 + 2 * half;
    float4 lo = make_float4(acc[0], acc[1], acc[2], acc[3]);
    float4 hi = make_float4(acc[4], acc[5], acc[6], acc[7]);
    *(float4*)(op + (size_t)ryo * OW + 2 * w0)       = lo;
    *(float4*)(op + (size_t)(ryo + 1) * OW + 2 * w0) = hi;
  }
}

extern "C" void kernel_launch(void* const* d_in, const int* in_sizes, int n_in,
                              void* d_out, int out_size, void* d_ws, size_t ws_size,
                              hipStream_t stream) {
  (void)in_sizes; (void)n_in; (void)out_size; (void)d_ws; (void)ws_size;
  const float* feat = (const float*)d_in[0];
  const float* mask = (const float*)d_in[1];
  float* out = (float*)d_out;
  dim3 grid(8 * 16, NG, 4);   // 128 super-tiles x 4 groups x 4 images
  carafe_wmma<<<grid, 256, 0, stream>>>(feat, mask, out);
}